// Block_6975026889363
// MI455X (gfx1250) — compile-verified
//
#include <hip/hip_runtime.h>
#include <math.h>

// ---------------------------------------------------------------------------
// Problem constants (from the reference): E=256, B=16, S=2048, fp32 everywhere
// ---------------------------------------------------------------------------
#define SEQ   2048
#define NBATCH 16
#define MTOK  (NBATCH * SEQ)   // 32768 tokens

typedef __attribute__((ext_vector_type(2))) float v2f;
typedef __attribute__((ext_vector_type(8))) float v8f;

__device__ __forceinline__ v8f wmma4(v2f a, v2f b, v8f c) {
  // V_WMMA_F32_16X16X4_F32 : D(16x16 f32) = A(16x4 f32) * B(4x16 f32) + C
  return __builtin_amdgcn_wmma_f32_16x16x4_f32(false, a, false, b, (short)0, c,
                                               false, false);
}
__device__ __forceinline__ float silu_f(float x) { return x / (1.0f + __expf(-x)); }
__device__ __forceinline__ float gelu_f(float x) {
  return 0.5f * x * (1.0f + erff(x * 0.7071067811865475f));  // exact GELU
}

// ---------------------------------------------------------------------------
// Kernel 1: t-vectors for the 4 temporal_linear layers + normalized time_embed
// grid = 5 blocks x 256 threads. blocks 0..3 -> layers, block 4 -> norm vector
// ---------------------------------------------------------------------------
struct TvecArgs {
  const float* l1w[4]; const float* l1b[4];
  const float* l2w[4]; const float* l2b[4];
  const float* te;
  float* tvec;   // [4][256]
  float* norm;   // [256]
};

__global__ __launch_bounds__(256) void tvec_norm_kernel(TvecArgs a) {
  __shared__ float shte[256];
  __shared__ float sh[256];
  const int tid = threadIdx.x;
  const int bb  = blockIdx.x;
  shte[tid] = a.te[tid];
  __syncthreads();
  if (bb == 4) {
    sh[tid] = shte[tid];
    __syncthreads();
    for (int s = 128; s > 0; s >>= 1) { if (tid < s) sh[tid] += sh[tid + s]; __syncthreads(); }
    const float mean = sh[0] * (1.0f / 256.0f);
    __syncthreads();
    const float d = shte[tid] - mean;
    sh[tid] = d * d;
    __syncthreads();
    for (int s = 128; s > 0; s >>= 1) { if (tid < s) sh[tid] += sh[tid + s]; __syncthreads(); }
    const float var = sh[0] * (1.0f / 256.0f);
    a.norm[tid] = (shte[tid] - mean) * rsqrtf(var + 1e-5f);
  } else {
    const float* w1 = a.l1w[bb];
    float acc = a.l1b[bb][tid];
    for (int k = 0; k < 256; ++k) acc += shte[k] * w1[k * 256 + tid];
    sh[tid] = silu_f(acc);
    __syncthreads();
    const float* w2 = a.l2w[bb];
    float acc2 = a.l2b[bb][tid];
    for (int k = 0; k < 256; ++k) acc2 += sh[k] * w2[k * 256 + tid];
    a.tvec[bb * 256 + tid] = acc2;
  }
}

// ---------------------------------------------------------------------------
// Kernel 2: fold the (buggy) temporal-LN into one affine map:
//   M[k,e] = fw_w[k,e]*norm[e] + fb_w[k,e];  c[e] = norm[e]*(fw_b[e]+1)+fb_b[e]
// grid = (256, 2) x 256 threads
// ---------------------------------------------------------------------------
struct CombArgs {
  const float* fww[2]; const float* fwb[2];
  const float* fbw[2]; const float* fbb[2];
  const float* norm;
  float* M[2]; float* c[2];
};

__global__ __launch_bounds__(256) void combine_kernel(CombArgs a) {
  const int set = blockIdx.y;
  const int idx = blockIdx.x * 256 + threadIdx.x;  // 0..65535 over [k][e]
  const int e   = idx & 255;
  const float nv = a.norm[e];
  a.M[set][idx] = a.fww[set][idx] * nv + a.fbw[set][idx];
  if (idx < 256) a.c[set][idx] = nv * (a.fwb[set][idx] + 1.0f) + a.fbb[set][idx];
}

// ---------------------------------------------------------------------------
// Kernel 3: weight generation matvec  out[N] = t(256) @ W(256xN) + bias[N]
// Streams up to 201 MB -> pure HBM bandwidth, coalesced float4.
// ---------------------------------------------------------------------------
__global__ __launch_bounds__(256) void matvec_kernel(const float* __restrict__ t,
                                                     const float* __restrict__ W,
                                                     const float* __restrict__ brow,
                                                     float* __restrict__ out, int N) {
  __shared__ float ts[256];
  const int tid = threadIdx.x;
  ts[tid] = t[tid];
  __syncthreads();
  const int col = (blockIdx.x * 256 + tid) * 4;
  if (col >= N) return;
  float4 acc = *(const float4*)(brow + col);
  for (int k = 0; k < 256; ++k) {
    const float tv = ts[k];
    const float4 w = *(const float4*)(W + (size_t)k * N + col);
    acc.x += tv * w.x; acc.y += tv * w.y; acc.z += tv * w.z; acc.w += tv * w.w;
  }
  *(float4*)(out + col) = acc;
}

// ---------------------------------------------------------------------------
// Kernel 4: fp32 WMMA GEMM  D[M,N] = act(A[M,256] @ W[256,N] + bias[N]) (+add)
// Block: 256 threads (8 waves), tile 128x128, waves 4x2 each computing 32x64.
// Double-buffered LDS: next K-slice is loaded into registers while computing
// the current one; one barrier per K-slice. 8 WMMA per 12 LDS float reads.
// LDS pads: As stride 20, Ws stride 132 -> conflict-free fragment reads,
// 16B-aligned float4 tile stores.
// ---------------------------------------------------------------------------
template <int ACT, bool ADD>   // ACT: 0=none 1=silu 2=gelu
__global__ __launch_bounds__(256) void gemm_wmma(const float* __restrict__ A,
                                                 const float* __restrict__ W,
                                                 const float* __restrict__ bias,
                                                 const float* __restrict__ addsrc,
                                                 float* __restrict__ D, int N) {
  __shared__ float As[2][128][20];
  __shared__ float Ws[2][16][132];
  const int tid  = threadIdx.x;
  const int wave = tid >> 5, lane = tid & 31;
  const int r    = lane & 15, half = lane >> 4;
  const int wm   = wave & 3,  wn   = wave >> 2;   // 4x2 waves, 32x64 each
  const int m_blk = blockIdx.y * 128;
  const int n_blk = blockIdx.x * 128;

  const v8f vzero = {0.f, 0.f, 0.f, 0.f, 0.f, 0.f, 0.f, 0.f};
  v8f c[2][4];
#pragma unroll
  for (int i = 0; i < 2; ++i)
#pragma unroll
    for (int j = 0; j < 4; ++j) c[i][j] = vzero;

  const int arow = tid >> 2;          // 0..63
  const int acol = (tid & 3) << 2;    // 0,4,8,12
  const int wrow = tid >> 4;          // 0..15
  const int wcol = (tid & 15) << 2;   // 0..60

  const float* aptr0 = A + (size_t)(m_blk + arow) * 256 + acol;
  const float* aptr1 = aptr0 + (size_t)64 * 256;
  const float* wptr  = W + (size_t)wrow * N + n_blk + wcol;

  // preload K-slice 0 and stage into buffer 0
  float4 va0 = *(const float4*)(aptr0);
  float4 va1 = *(const float4*)(aptr1);
  float4 vw0 = *(const float4*)(wptr);
  float4 vw1 = *(const float4*)(wptr + 64);
  *(float4*)&As[0][arow][acol]      = va0;
  *(float4*)&As[0][arow + 64][acol] = va1;
  *(float4*)&Ws[0][wrow][wcol]      = vw0;
  *(float4*)&Ws[0][wrow][wcol + 64] = vw1;
  __syncthreads();

  for (int kc = 0; kc < 16; ++kc) {
    const int cur = kc & 1;
    if (kc < 15) {  // prefetch next K-slice into registers
      const size_t koff = (size_t)(kc + 1) * 16;
      va0 = *(const float4*)(aptr0 + koff);
      va1 = *(const float4*)(aptr1 + koff);
      vw0 = *(const float4*)(wptr + koff * N);
      vw1 = *(const float4*)(wptr + koff * N + 64);
    }
#pragma unroll
    for (int kk = 0; kk < 16; kk += 4) {
      const int k0 = kk + 2 * half;   // VGPR0 holds K=k0, VGPR1 holds K=k0+1
      v2f a0, a1, b[4];
      a0.x = As[cur][wm * 32 + r][k0];       a0.y = As[cur][wm * 32 + r][k0 + 1];
      a1.x = As[cur][wm * 32 + 16 + r][k0];  a1.y = As[cur][wm * 32 + 16 + r][k0 + 1];
#pragma unroll
      for (int j = 0; j < 4; ++j) {
        b[j].x = Ws[cur][k0][wn * 64 + j * 16 + r];
        b[j].y = Ws[cur][k0 + 1][wn * 64 + j * 16 + r];
      }
#pragma unroll
      for (int j = 0; j < 4; ++j) {
        c[0][j] = wmma4(a0, b[j], c[0][j]);
        c[1][j] = wmma4(a1, b[j], c[1][j]);
      }
    }
    if (kc < 15) {  // stage prefetched slice into the other buffer
      const int nxt = cur ^ 1;
      *(float4*)&As[nxt][arow][acol]      = va0;
      *(float4*)&As[nxt][arow + 64][acol] = va1;
      *(float4*)&Ws[nxt][wrow][wcol]      = vw0;
      *(float4*)&Ws[nxt][wrow][wcol + 64] = vw1;
    }
    __syncthreads();
  }

#pragma unroll
  for (int i = 0; i < 2; ++i)
#pragma unroll
    for (int j = 0; j < 4; ++j) {
      const int gc = n_blk + wn * 64 + j * 16 + r;
      const float bv = bias[gc];
#pragma unroll
      for (int v = 0; v < 8; ++v) {
        const int gr = m_blk + wm * 32 + i * 16 + v + 8 * half;
        float val = c[i][j][v] + bv;
        if (ACT == 1) val = silu_f(val);
        else if (ACT == 2) val = gelu_f(val);
        const size_t idx = (size_t)gr * (size_t)N + gc;
        if (ADD) val += addsrc[idx];
        D[idx] = val;
      }
    }
}

// ---------------------------------------------------------------------------
// Kernel 5: flash attention, 1 head, d=256, scale 1/16.
// Block = 128 threads (4 waves), 64 query rows/block, 32-key tiles.
// QK^T and P@V via f32 WMMA; online softmax with 16-lane shfl reductions.
// ---------------------------------------------------------------------------
__global__ __launch_bounds__(128) void fa_kernel(const float* __restrict__ qkv,
                                                 float* __restrict__ out) {
  __shared__ float Qs[64][260];
  __shared__ float Ks[32][260];
  __shared__ float Vs[32][260];
  __shared__ float Ps[4][16][36];

  const int b   = blockIdx.y;
  const int q0  = blockIdx.x * 64;
  const int tid = threadIdx.x;
  const int wave = tid >> 5, lane = tid & 31;
  const int r = lane & 15, half = lane >> 4;

  const float* qbase = qkv + ((size_t)b * SEQ + q0) * 768;
  for (int i = tid; i < 64 * 64; i += 128) {
    const int row = i >> 6;
    const int cg  = (i & 63) << 2;
    const float4 v = *(const float4*)(qbase + (size_t)row * 768 + cg);
    Qs[row][cg + 0] = v.x * 0.0625f;  // fold 1/sqrt(256) into Q
    Qs[row][cg + 1] = v.y * 0.0625f;
    Qs[row][cg + 2] = v.z * 0.0625f;
    Qs[row][cg + 3] = v.w * 0.0625f;
  }

  const v8f vzero = {0.f, 0.f, 0.f, 0.f, 0.f, 0.f, 0.f, 0.f};
  v8f o_acc[16];
#pragma unroll
  for (int n = 0; n < 16; ++n) o_acc[n] = vzero;
  float m_i[8], l_i[8];
#pragma unroll
  for (int v = 0; v < 8; ++v) { m_i[v] = -1e30f; l_i[v] = 0.0f; }

  const int qrow = wave * 16 + r;

  for (int kt = 0; kt < SEQ / 32; ++kt) {
    __syncthreads();
    const float* kbase = qkv + ((size_t)b * SEQ + kt * 32) * 768 + 256;
    for (int i = tid; i < 32 * 64; i += 128) {
      const int row = i >> 6;
      const int cg  = (i & 63) << 2;
      *(float4*)&Ks[row][cg] = *(const float4*)(kbase + (size_t)row * 768 + cg);
      *(float4*)&Vs[row][cg] = *(const float4*)(kbase + (size_t)row * 768 + 256 + cg);
    }
    __syncthreads();

    // S(16x32) = Qw(16x256) @ K^T(256x32)
    v8f s0 = vzero, s1 = vzero;
    for (int kk = 0; kk < 256; kk += 4) {
      const int k0 = kk + 2 * half;
      v2f aq; aq.x = Qs[qrow][k0];       aq.y = Qs[qrow][k0 + 1];
      v2f b0; b0.x = Ks[r][k0];          b0.y = Ks[r][k0 + 1];
      v2f b1; b1.x = Ks[16 + r][k0];     b1.y = Ks[16 + r][k0 + 1];
      s0 = wmma4(aq, b0, s0);
      s1 = wmma4(aq, b1, s1);
    }

    // online softmax (row v+8*half of the wave's 16-row tile)
#pragma unroll
    for (int v = 0; v < 8; ++v) {
      float tmax = fmaxf(s0[v], s1[v]);
      for (int off = 1; off < 16; off <<= 1)
        tmax = fmaxf(tmax, __shfl_xor(tmax, off, 32));
      const float mn = fmaxf(m_i[v], tmax);
      const float alpha = __expf(m_i[v] - mn);
      const float p0 = __expf(s0[v] - mn);
      const float p1 = __expf(s1[v] - mn);
      float tsum = p0 + p1;
      for (int off = 1; off < 16; off <<= 1)
        tsum += __shfl_xor(tsum, off, 32);
      l_i[v] = l_i[v] * alpha + tsum;
      m_i[v] = mn;
#pragma unroll
      for (int n = 0; n < 16; ++n) o_acc[n][v] *= alpha;
      Ps[wave][v + 8 * half][r]      = p0;   // C-layout -> A-layout via LDS
      Ps[wave][v + 8 * half][16 + r] = p1;
    }

    // O(16x256) += P(16x32) @ V(32x256)
    for (int kk = 0; kk < 32; kk += 4) {
      const int k0 = kk + 2 * half;
      v2f ap; ap.x = Ps[wave][r][k0]; ap.y = Ps[wave][r][k0 + 1];
#pragma unroll
      for (int n = 0; n < 16; ++n) {
        v2f bv; bv.x = Vs[k0][n * 16 + r]; bv.y = Vs[k0 + 1][n * 16 + r];
        o_acc[n] = wmma4(ap, bv, o_acc[n]);
      }
    }
  }

  float* obase = out + ((size_t)b * SEQ + q0 + wave * 16) * 256;
#pragma unroll
  for (int n = 0; n < 16; ++n)
#pragma unroll
    for (int v = 0; v < 8; ++v) {
      const int row = v + 8 * half;
      const int col = n * 16 + r;
      obase[(size_t)row * 256 + col] = o_acc[n][v] * (1.0f / l_i[v]);
    }
}

// ---------------------------------------------------------------------------
// Host-side orchestration
// ---------------------------------------------------------------------------
extern "C" void kernel_launch(void* const* d_in, const int* in_sizes, int n_in,
                              void* d_out, int out_size, void* d_ws, size_t ws_size,
                              hipStream_t stream) {
  (void)n_in; (void)out_size; (void)ws_size;
  auto F = [&](int i) { return (const float*)d_in[i]; };

  // Input flattening: detect insertion-order (in_sizes[0]==65536, aln.lin1_w)
  // vs jax-pytree sorted keys (in_sizes[0]==256, aln.fb_b).
  const bool sorted = (in_sizes[0] == 256);

  const float *aln_l1w, *aln_l1b, *aln_l2w, *aln_l2b, *aln_fww, *aln_fwb, *aln_fbw, *aln_fbb;
  const float *mln_l1w, *mln_l1b, *mln_l2w, *mln_l2b, *mln_fww, *mln_fwb, *mln_fbw, *mln_fbb;
  const float *cat_l1w, *cat_l1b, *cat_l2w, *cat_l2b, *cat_fWw, *cat_fWb, *cat_fbw, *cat_fbb;
  const float *prj_l1w, *prj_l1b, *prj_l2w, *prj_l2b, *prj_fWw, *prj_fWb, *prj_fbw, *prj_fbb;
  const float *cfc_l1w, *cfc_l1b, *cfc_l2w, *cfc_l2b, *cfc_fWw, *cfc_fWb, *cfc_fbw, *cfc_fbb;
  const float *mlp_l1w, *mlp_l1b, *mlp_l2w, *mlp_l2b, *mlp_fWw, *mlp_fWb, *mlp_fbw, *mlp_fbb;
  const float *te, *x;

  if (!sorted) {
    // recursive insertion order: p_aln, p_mln, p_cattn, p_cproj, p_cfc, p_cproj_mlp, te, x
    aln_l1w=F(0);  aln_l1b=F(1);  aln_l2w=F(2);  aln_l2b=F(3);
    aln_fww=F(4);  aln_fwb=F(5);  aln_fbw=F(6);  aln_fbb=F(7);
    mln_l1w=F(8);  mln_l1b=F(9);  mln_l2w=F(10); mln_l2b=F(11);
    mln_fww=F(12); mln_fwb=F(13); mln_fbw=F(14); mln_fbb=F(15);
    cat_l1w=F(16); cat_l1b=F(17); cat_l2w=F(18); cat_l2b=F(19);
    cat_fWw=F(20); cat_fWb=F(21); cat_fbw=F(22); cat_fbb=F(23);
    prj_l1w=F(24); prj_l1b=F(25); prj_l2w=F(26); prj_l2b=F(27);
    prj_fWw=F(28); prj_fWb=F(29); prj_fbw=F(30); prj_fbb=F(31);
    cfc_l1w=F(32); cfc_l1b=F(33); cfc_l2w=F(34); cfc_l2b=F(35);
    cfc_fWw=F(36); cfc_fWb=F(37); cfc_fbw=F(38); cfc_fbb=F(39);
    mlp_l1w=F(40); mlp_l1b=F(41); mlp_l2w=F(42); mlp_l2b=F(43);
    mlp_fWw=F(44); mlp_fWb=F(45); mlp_fbw=F(46); mlp_fbb=F(47);
    te=F(48); x=F(49);
  } else {
    // pytree sorted: p_aln(0), p_cattn(8), p_cfc(16), p_cproj(24), p_cproj_mlp(32), p_mln(40)
    // TLN keys sorted: fb_b, fb_w, fw_b, fw_w, lin1_b, lin1_w, lin2_b, lin2_w
    // TL  keys sorted: fW_b, fW_w, fb_b, fb_w, lin1_b, lin1_w, lin2_b, lin2_w
    aln_fbb=F(0);  aln_fbw=F(1);  aln_fwb=F(2);  aln_fww=F(3);
    aln_l1b=F(4);  aln_l1w=F(5);  aln_l2b=F(6);  aln_l2w=F(7);
    cat_fWb=F(8);  cat_fWw=F(9);  cat_fbb=F(10); cat_fbw=F(11);
    cat_l1b=F(12); cat_l1w=F(13); cat_l2b=F(14); cat_l2w=F(15);
    cfc_fWb=F(16); cfc_fWw=F(17); cfc_fbb=F(18); cfc_fbw=F(19);
    cfc_l1b=F(20); cfc_l1w=F(21); cfc_l2b=F(22); cfc_l2w=F(23);
    prj_fWb=F(24); prj_fWw=F(25); prj_fbb=F(26); prj_fbw=F(27);
    prj_l1b=F(28); prj_l1w=F(29); prj_l2b=F(30); prj_l2w=F(31);
    mlp_fWb=F(32); mlp_fWw=F(33); mlp_fbb=F(34); mlp_fbw=F(35);
    mlp_l1b=F(36); mlp_l1w=F(37); mlp_l2b=F(38); mlp_l2w=F(39);
    mln_fbb=F(40); mln_fbw=F(41); mln_fwb=F(42); mln_fww=F(43);
    mln_l1b=F(44); mln_l1w=F(45); mln_l2b=F(46); mln_l2w=F(47);
    te=F(48); x=F(49);
  }

  // ---- workspace layout (floats) ----
  float* ws = (float*)d_ws;
  size_t off = 0;
  auto alloc = [&](size_t n) { float* p = ws + off; off += n; return p; };
  float* norm   = alloc(256);
  float* tvec   = alloc(4 * 256);     // [cattn, cproj, cfc, cproj_mlp]
  float* M_aln  = alloc(65536); float* c_aln = alloc(256);
  float* M_mln  = alloc(65536); float* c_mln = alloc(256);
  float* W_qkv  = alloc(196608); float* b_qkv = alloc(768);
  float* W_proj = alloc(65536);  float* b_proj = alloc(256);
  float* W_fc   = alloc(65536);  float* b_fc   = alloc(256);
  float* W_mlp  = alloc(65536);  float* b_mlp  = alloc(256);
  float* h1     = alloc((size_t)MTOK * 256);
  float* tbuf   = alloc((size_t)MTOK * 256);
  float* ln_a   = alloc((size_t)MTOK * 256);
  float* ln_m   = alloc((size_t)MTOK * 256);
  float* qkvb   = alloc((size_t)MTOK * 768);
  float* attn_out = h1;     // reuse (h1 free after TLN-mln)
  float* a_branch = ln_a;   // reuse (ln_a free after qkv GEMM)
  float* h_mlp    = tbuf;   // reuse (tbuf free after TLN-mln)

  // 1) t-vectors + norm vector
  TvecArgs ta;
  ta.l1w[0]=cat_l1w; ta.l1b[0]=cat_l1b; ta.l2w[0]=cat_l2w; ta.l2b[0]=cat_l2b;
  ta.l1w[1]=prj_l1w; ta.l1b[1]=prj_l1b; ta.l2w[1]=prj_l2w; ta.l2b[1]=prj_l2b;
  ta.l1w[2]=cfc_l1w; ta.l1b[2]=cfc_l1b; ta.l2w[2]=cfc_l2w; ta.l2b[2]=cfc_l2b;
  ta.l1w[3]=mlp_l1w; ta.l1b[3]=mlp_l1b; ta.l2w[3]=mlp_l2w; ta.l2b[3]=mlp_l2b;
  ta.te = te; ta.tvec = tvec; ta.norm = norm;
  tvec_norm_kernel<<<dim3(5), dim3(256), 0, stream>>>(ta);

  // 2) fold LN into affine maps
  CombArgs ca;
  ca.fww[0]=aln_fww; ca.fwb[0]=aln_fwb; ca.fbw[0]=aln_fbw; ca.fbb[0]=aln_fbb;
  ca.fww[1]=mln_fww; ca.fwb[1]=mln_fwb; ca.fbw[1]=mln_fbw; ca.fbb[1]=mln_fbb;
  ca.norm = norm; ca.M[0]=M_aln; ca.c[0]=c_aln; ca.M[1]=M_mln; ca.c[1]=c_mln;
  combine_kernel<<<dim3(256, 2), dim3(256), 0, stream>>>(ca);

  // 3) generate data-dependent weights/biases (bandwidth-bound matvecs)
  auto mv = [&](const float* t, const float* W, const float* b, float* o, int N) {
    const int blocks = (N / 4 + 255) / 256;
    matvec_kernel<<<dim3(blocks), dim3(256), 0, stream>>>(t, W, b, o, N);
  };
  mv(tvec + 0,   cat_fWw, cat_fWb, W_qkv, 196608);
  mv(tvec + 0,   cat_fbw, cat_fbb, b_qkv, 768);
  mv(tvec + 256, prj_fWw, prj_fWb, W_proj, 65536);
  mv(tvec + 256, prj_fbw, prj_fbb, b_proj, 256);
  mv(tvec + 512, cfc_fWw, cfc_fWb, W_fc, 65536);
  mv(tvec + 512, cfc_fbw, cfc_fbb, b_fc, 256);
  mv(tvec + 768, mlp_fWw, mlp_fWb, W_mlp, 65536);
  mv(tvec + 768, mlp_fbw, mlp_fbb, b_mlp, 256);

  const dim3 g256(256 / 128, MTOK / 128), g768(768 / 128, MTOK / 128), blk(256);

  // 4) TLN (attention branch): x -> silu -> lin2 -> affine LN
  gemm_wmma<1, false><<<g256, blk, 0, stream>>>(x,    aln_l1w, aln_l1b, nullptr, h1,   256);
  gemm_wmma<0, false><<<g256, blk, 0, stream>>>(h1,   aln_l2w, aln_l2b, nullptr, tbuf, 256);
  gemm_wmma<0, false><<<g256, blk, 0, stream>>>(tbuf, M_aln,   c_aln,   nullptr, ln_a, 256);
  // TLN (mlp branch)
  gemm_wmma<1, false><<<g256, blk, 0, stream>>>(x,    mln_l1w, mln_l1b, nullptr, h1,   256);
  gemm_wmma<0, false><<<g256, blk, 0, stream>>>(h1,   mln_l2w, mln_l2b, nullptr, tbuf, 256);
  gemm_wmma<0, false><<<g256, blk, 0, stream>>>(tbuf, M_mln,   c_mln,   nullptr, ln_m, 256);

  // 5) QKV projection and attention
  gemm_wmma<0, false><<<g768, blk, 0, stream>>>(ln_a, W_qkv, b_qkv, nullptr, qkvb, 768);
  fa_kernel<<<dim3(SEQ / 64, NBATCH), dim3(128), 0, stream>>>(qkvb, attn_out);

  // 6) output projection of attention branch
  gemm_wmma<0, false><<<g256, blk, 0, stream>>>(attn_out, W_proj, b_proj, nullptr, a_branch, 256);

  // 7) MLP branch: gelu(ln_m @ W_fc + b) @ W_mlp + b + a_branch -> d_out
  gemm_wmma<2, false><<<g256, blk, 0, stream>>>(ln_m,  W_fc,  b_fc,  nullptr,  h_mlp, 256);
  gemm_wmma<0, true ><<<g256, blk, 0, stream>>>(h_mlp, W_mlp, b_mlp, a_branch, (float*)d_out, 256);
}